// DistributedSpectralConvS2_67259187855991
// MI455X (gfx1250) — compile-verified
//
#include <hip/hip_runtime.h>
#include <hip/hip_bf16.h>

// ---------------------------------------------------------------------------
// SpectralConvS2 on MI455X (gfx1250, wave32), all-WMMA pipeline:
//   A: xf = x * DFT            [16384x512]x[512x272]     (re,im)   WN=1 BV=1
//   P: permute [c][k][m] -> [m][c][k]  (LDS tile transpose, k-contiguous B)
//   B: coeffs_m = sht_w[m]*xf_m  257 x [256x256]x[256x64]          WN=4 BV=1
//   C: out_l = coeffs_l * W_l    256 x complex dual-GEMM [272x64]x[64x64]
//   D: xk_m = out_m * pct[m]     257 x [64x256]x[256x256]          WN=4 BV=0
//   E: y = xk_r*Gr + xk_i*Gi     [16384x272]x[272x512] dual -> d_out
//   + residual copy.
// fp32 WMMA (V_WMMA_F32_16X16X4_F32): SHT needs f32 accuracy; ~28 GFLOP vs
// ~220MB HBM traffic (~9.5us @ 23.3 TB/s) makes precision, not TOPS, binding.
// All M,N padded to multiples of 16 -> zero bounds checks, full-EXEC waves,
// b64 fragment loads, pointer-increment addressing, A-fragment reuse (WN=4).
// ---------------------------------------------------------------------------

typedef float v2f __attribute__((ext_vector_type(2)));
typedef float v8f __attribute__((ext_vector_type(8)));

#define NLAT 256
#define NLON 512
#define MMAX 257
#define MPAD 272          // 257 padded to multiple of 16
#define CIN  64
#define COUT 64
#define LMAX 256
#define RROWS (CIN * NLAT)   // 16384 (c,k) rows

// ---------------------------------------------------------------------------
// Batched strided WMMA GEMM. Wave computes a 16 x (WN*16) tile; 4 waves/block
// cover 4 row tiles. DUAL: D = A1*B1 + A2*B2 (complex products; the minus is
// pre-folded into the B2 operand since f32 WMMA has no A/B NEG per ISA).
// BV: B operand is k-contiguous (brs==1) -> b64 loads. A is always acs==1.
// Grid exactly tiles N; only row tiles are guarded (wave-uniform).
// ---------------------------------------------------------------------------
template<int WN, int DUAL, int BV>
__global__ void __launch_bounds__(128) gemmT(
    const float* __restrict__ A,  const float* __restrict__ A2,
    const float* __restrict__ B,  const float* __restrict__ B2,
    float* __restrict__ C,
    int M, int K,
    long long ars, long long abat,
    long long brs, long long bcs, long long bbat,
    long long crs, long long ccs, long long cbat)
{
    const int wave = threadIdx.x >> 5;
    const int lane = threadIdx.x & 31;
    const int half = lane >> 4;     // 0: lanes 0-15, 1: lanes 16-31
    const int l16  = lane & 15;

    const int row0 = (blockIdx.y * 4 + wave) * 16;
    if (row0 >= M) return;                    // wave-uniform: EXEC stays ~0
    const int col0 = blockIdx.x * (WN * 16);

    // ISA 7.12.2 (wave32): A frag: m=lane%16, k=2*(lane/16)+{0,1}
    //                      B frag: n=lane%16, k=2*(lane/16)+{0,1}
    //                      C frag: row=j+8*(lane/16), col=lane%16
    const long long z    = blockIdx.z;
    const long long aofs = z * abat + (long long)(row0 + l16) * ars + 2 * half;
    const float* Ap  = A  + aofs;
    const float* A2p = A2 + aofs;

    const float* Bp[WN];
    const float* B2p[WN];
#pragma unroll
    for (int t = 0; t < WN; ++t) {
        const long long bofs = z * bbat + (long long)(col0 + t * 16 + l16) * bcs
                             + (long long)(2 * half) * brs;
        Bp[t]  = B  + bofs;
        B2p[t] = B2 + bofs;
    }

    v8f acc[WN];
#pragma unroll
    for (int t = 0; t < WN; ++t) acc[t] = (v8f){0.f,0.f,0.f,0.f,0.f,0.f,0.f,0.f};

    const long long bstep = BV ? 4 : 4 * brs;

#pragma unroll 2
    for (int k0 = 0; k0 < K; k0 += 4) {
        v2f a = *(const v2f*)Ap;  Ap += 4;
        v2f a2 = a;
        if (DUAL) { a2 = *(const v2f*)A2p; A2p += 4; }
#pragma unroll
        for (int t = 0; t < WN; ++t) {
            v2f b;
            if (BV) { b = *(const v2f*)Bp[t]; }
            else    { b.x = Bp[t][0]; b.y = Bp[t][brs]; }
            Bp[t] += bstep;
            acc[t] = __builtin_amdgcn_wmma_f32_16x16x4_f32(
                         false, a, false, b, (short)0, acc[t], false, false);
            if (DUAL) {
                v2f b2;
                if (BV) { b2 = *(const v2f*)B2p[t]; }
                else    { b2.x = B2p[t][0]; b2.y = B2p[t][brs]; }
                B2p[t] += bstep;
                acc[t] = __builtin_amdgcn_wmma_f32_16x16x4_f32(
                             false, a2, false, b2, (short)0, acc[t], false, false);
            }
        }
    }

    const long long cbase = z * cbat + (long long)(col0 + l16) * ccs;
#pragma unroll
    for (int t = 0; t < WN; ++t) {
#pragma unroll
        for (int j = 0; j < 8; ++j) {
            const long long r = row0 + j + 8 * half;
            C[cbase + (long long)t * 16 * ccs + r * crs] = acc[t][j];
        }
    }
}

// ---------------------------------------------------------------------------
// DFT tables. Forward stored [m=272][n=512] (k-contiguous for BV=1 B loads):
//   FrT[m][n] = (2pi/512) cos(2pi mn/512), FiT = -(2pi/512) sin(.)
// Inverse stored [n=512][m=272] (k=m contiguous):
//   GrT[n][m] = c_m cos(2pi mn/512), GiT = -c_m sin(.), c_0=c_256=1 else 2.
// Pad entries m>=257 are exact zeros.
// ---------------------------------------------------------------------------
__global__ void build_tables(float* __restrict__ FrT, float* __restrict__ FiT,
                             float* __restrict__ GrT, float* __restrict__ GiT)
{
    const int idx = blockIdx.x * blockDim.x + threadIdx.x;
    if (idx >= MPAD * NLON) return;
    const float two_pi = 6.28318530717958647692f;
    const float step   = two_pi / (float)NLON;

    {   // forward tables, [m][n]
        const int m = idx / NLON, n = idx % NLON;
        float vr = 0.f, vi = 0.f;
        if (m < MMAX) {
            const int ph = (int)(((long long)m * (long long)n) & (NLON - 1));
            const float ang = step * (float)ph;
            vr =  step * cosf(ang);
            vi = -step * sinf(ang);
        }
        FrT[idx] = vr; FiT[idx] = vi;
    }
    {   // inverse tables, [n][m]
        const int n = idx / MPAD, m = idx % MPAD;
        float vr = 0.f, vi = 0.f;
        if (m < MMAX) {
            const int ph = (int)(((long long)m * (long long)n) & (NLON - 1));
            const float ang = step * (float)ph;
            const float cm  = (m == 0 || m == MMAX - 1) ? 1.f : 2.f;
            vr =  cm * cosf(ang);
            vi = -cm * sinf(ang);
        }
        GrT[idx] = vr; GiT[idx] = vi;
    }
}

// [g=1][i][o][l] -> [l][o][i] (i-contiguous = k-contiguous for stage C),
// plus pre-negated imag for the complex product's minus term.
__global__ void permute_weights(const float* __restrict__ wr,
                                const float* __restrict__ wi,
                                float* __restrict__ wlr,
                                float* __restrict__ wli,
                                float* __restrict__ wlni)
{
    const int idx = blockIdx.x * blockDim.x + threadIdx.x;
    if (idx >= LMAX * CIN * COUT) return;
    const int i = idx & 63;
    const int o = (idx >> 6) & 63;
    const int l = idx >> 12;
    const size_t s = ((size_t)i * COUT + o) * LMAX + l;
    const float r  = wr[s];
    const float im = wi[s];
    wlr[idx]  = r;
    wli[idx]  = im;
    wlni[idx] = -im;
}

// xf [c][k][MPAD] -> xfp [m][c][k] via 16x16 LDS tile transpose over (k,m).
__global__ void __launch_bounds__(256) permute_ckm_to_mck(
    const float* __restrict__ src, float* __restrict__ dst)
{
    __shared__ float tile[16][17];   // padded: conflict-free transpose
    const int c  = blockIdx.z;
    const int m0 = blockIdx.x * 16;
    const int k0 = blockIdx.y * 16;
    const int tx = threadIdx.x, ty = threadIdx.y;
    tile[ty][tx] = src[((size_t)c * NLAT + (k0 + ty)) * MPAD + (m0 + tx)];
    __syncthreads();
    dst[((size_t)(m0 + ty) * CIN + c) * NLAT + (k0 + tx)] = tile[tx][ty];
}

__global__ void zero_fill(float* __restrict__ p, size_t n)
{
    size_t i = (size_t)blockIdx.x * blockDim.x + threadIdx.x;
    const size_t stride = (size_t)gridDim.x * blockDim.x;
    for (; i < n; i += stride) p[i] = 0.f;
}

__global__ void copy_f4(const float4* __restrict__ s, float4* __restrict__ d, int n)
{
    const int i = blockIdx.x * blockDim.x + threadIdx.x;
    if (i < n) d[i] = s[i];
}

// ---------------------------------------------------------------------------
extern "C" void kernel_launch(void* const* d_in, const int* in_sizes, int n_in,
                              void* d_out, int out_size, void* d_ws, size_t ws_size,
                              hipStream_t stream)
{
    (void)in_sizes; (void)n_in; (void)out_size;
    const float* x     = (const float*)d_in[0];   // [1,64,256,512]
    const float* w_r   = (const float*)d_in[1];   // [1,64,64,256]
    const float* w_i   = (const float*)d_in[2];   // [1,64,64,256]
    const float* pct   = (const float*)d_in[3];   // [257,256,256]
    const float* sht_w = (const float*)d_in[4];   // [257,256,256]
    float* y    = (float*)d_out;                  // [1,64,256,512]
    float* resi = (float*)d_out + (size_t)RROWS * NLON;

    // workspace layout (floats)
    const size_t TAB = (size_t)MPAD * NLON;          // 139,264
    const size_t WSZ = (size_t)LMAX * CIN * COUT;    // 1,048,576
    const size_t BSZ = (size_t)MPAD * NLAT * CIN;    // 4,456,448 (= 16384*272)
    float* ws  = (float*)d_ws;
    float* FRT = ws;                 // [272][512]
    float* FIT = FRT + TAB;
    float* GRT = FIT + TAB;          // [512][272]
    float* GIT = GRT + TAB;
    float* WLR = GIT + TAB;          // [l][o][i]
    float* WLI = WLR + WSZ;
    float* WLN = WLI + WSZ;          // -imag
    float* B0  = WLN + WSZ;          // ping-pong buffers
    float* B1  = B0  + BSZ;
    float* B2  = B1  + BSZ;
    float* B3  = B2  + BSZ;
    const size_t need = (size_t)(B3 + BSZ - ws) * sizeof(float);
    if (ws_size < need) return;

    const dim3 blk(128);

    // 0) tables + weight permutation
    build_tables<<<dim3((MPAD * NLON + 255) / 256), dim3(256), 0, stream>>>(
        FRT, FIT, GRT, GIT);
    permute_weights<<<dim3((LMAX * CIN * COUT + 255) / 256), dim3(256), 0, stream>>>(
        w_r, w_i, WLR, WLI, WLN);

    // A) forward DFT: [16384x512] x [512x272] -> B0 (re), B1 (im). B=[m][n].
    {
        dim3 g(MPAD / 16, RROWS / 64, 1);
        gemmT<1, 0, 1><<<g, blk, 0, stream>>>(x, x, FRT, FRT, B0,
            RROWS, NLON,
            NLON, 0,  1, NLON, 0,  MPAD, 1, 0);
        gemmT<1, 0, 1><<<g, blk, 0, stream>>>(x, x, FIT, FIT, B1,
            RROWS, NLON,
            NLON, 0,  1, NLON, 0,  MPAD, 1, 0);
    }

    // P) [c][k][m] -> [m][c][k]: B0->B2, B1->B3
    {
        dim3 g(MPAD / 16, NLAT / 16, CIN), b(16, 16);
        permute_ckm_to_mck<<<g, b, 0, stream>>>(B0, B2);
        permute_ckm_to_mck<<<g, b, 0, stream>>>(B1, B3);
    }

    // B) forward Legendre: 257 batches sht_w[m][256x256] x xf_m[256x64].
    //    coeffs [m][l][c] -> B0 (re), B1 (im). B=[m][c][k].
    {
        dim3 g(1, LMAX / 64, MMAX);
        gemmT<4, 0, 1><<<g, blk, 0, stream>>>(sht_w, sht_w, B2, B2, B0,
            LMAX, NLAT,
            NLAT, (long long)LMAX * NLAT,
            1, NLAT, (long long)CIN * NLAT,
            CIN, 1, (long long)LMAX * CIN);
        gemmT<4, 0, 1><<<g, blk, 0, stream>>>(sht_w, sht_w, B3, B3, B1,
            LMAX, NLAT,
            NLAT, (long long)LMAX * NLAT,
            1, NLAT, (long long)CIN * NLAT,
            CIN, 1, (long long)LMAX * CIN);
    }

    // C) complex channel mix: 256 l-batches, M padded 257->272 (extra rows
    //    read in-workspace garbage and write rows stage D never reads).
    //    out stored [m][o][l] -> B2 (re), B3 (im). B=[l][o][i].
    {
        dim3 g(1, MPAD / 64 + 1, LMAX);   // 5 row blocks cover 272 rows
        // out_r = coeffs_r*Wr + coeffs_i*(-Wi)
        gemmT<4, 1, 1><<<g, blk, 0, stream>>>(B0, B1, WLR, WLN, B2,
            MPAD, CIN,
            (long long)LMAX * CIN, CIN,
            1, CIN, (long long)CIN * COUT,
            (long long)COUT * LMAX, LMAX, 1);
        // out_i = coeffs_r*Wi + coeffs_i*Wr
        gemmT<4, 1, 1><<<g, blk, 0, stream>>>(B0, B1, WLI, WLR, B3,
            MPAD, CIN,
            (long long)LMAX * CIN, CIN,
            1, CIN, (long long)CIN * COUT,
            (long long)COUT * LMAX, LMAX, 1);
    }

    // xk buffers must be zero in the m-pad columns (257..271)
    zero_fill<<<dim3(2048), dim3(256), 0, stream>>>(B0, BSZ);
    zero_fill<<<dim3(2048), dim3(256), 0, stream>>>(B1, BSZ);

    // D) inverse Legendre: 257 batches out_m[64x256] x pct[m][256x256].
    //    xk stored [o][k][MPAD] with per-batch column offset m -> B0, B1.
    //    B = pct as given (strided b32 loads; 67MB, L2-resident).
    {
        dim3 g(NLAT / 64, 1, MMAX);
        gemmT<4, 0, 0><<<g, blk, 0, stream>>>(B2, B2, pct, pct, B0,
            COUT, LMAX,
            LMAX, (long long)COUT * LMAX,
            NLAT, 1, (long long)LMAX * NLAT,
            (long long)NLAT * MPAD, MPAD, 1);
        gemmT<4, 0, 0><<<g, blk, 0, stream>>>(B3, B3, pct, pct, B1,
            COUT, LMAX,
            LMAX, (long long)COUT * LMAX,
            NLAT, 1, (long long)LMAX * NLAT,
            (long long)NLAT * MPAD, MPAD, 1);
    }

    // E) inverse DFT (dual): y = xk_r*Gr + xk_i*Gi, [16384x272]x[272x512].
    //    B = GrT/GiT stored [n][m] (k=m contiguous).
    {
        dim3 g(NLON / 64, RROWS / 64, 1);
        gemmT<4, 1, 1><<<g, blk, 0, stream>>>(B0, B1, GRT, GIT, y,
            RROWS, MPAD,
            MPAD, 0,
            1, MPAD, 0,
            NLON, 1, 0);
    }

    // residual = x
    {
        const int n4 = (RROWS * NLON) / 4;   // 2,097,152 float4
        copy_f4<<<dim3((n4 + 255) / 256), dim3(256), 0, stream>>>(
            (const float4*)x, (float4*)resi, n4);
    }
}